// GATEncoder_22574348108302
// MI455X (gfx1250) — compile-verified
//
#include <hip/hip_runtime.h>
#include <math.h>

typedef __attribute__((ext_vector_type(2))) float v2f;
typedef __attribute__((ext_vector_type(8))) float v8f;

#define IN_CH 256
#define HC    128   // H*C
#define HEADS 4
#define CPH   32    // channels per head
#define OUTC  64
#define NEG_SLOPE 0.2f

// ---------------- init: zero agg/denom, emax = -inf --------------------------
__global__ void k_init(float* __restrict__ agg, float* __restrict__ emax,
                       float* __restrict__ denom, int n) {
    int i = blockIdx.x * blockDim.x + threadIdx.x;
    if (i < n * HC) agg[i] = 0.0f;
    if (i < n * HEADS) { emax[i] = -INFINITY; denom[i] = 0.0f; }
}

// ---------------- GEMM1: h = x @ W  (fp32 WMMA 16x16x4) ----------------------
// block = 256 threads = 8 waves; wave w computes tile rows[b*16..+16) x cols[w*16..+16)
__global__ __launch_bounds__(256) void k_gemm_xw(const float* __restrict__ x,
                                                 const float* __restrict__ W,
                                                 float* __restrict__ h) {
    const int wave = threadIdx.x >> 5;
    const int lane = threadIdx.x & 31;
    const int row0 = blockIdx.x * 16;
    const int col0 = wave * 16;
    const int m  = lane & 15;          // tile row (A) / tile col (B)
    const int kh = (lane >> 4) << 1;   // 0 or 2: K sub-offset for this half-wave

    const float* __restrict__ xrow = x + (size_t)(row0 + m) * IN_CH;
    v8f acc = {};
    #pragma unroll 4
    for (int k = 0; k < IN_CH; k += 4) {
        v2f a, b;
        a.x = xrow[k + kh];
        a.y = xrow[k + kh + 1];
        b.x = W[(size_t)(k + kh)     * HC + col0 + m];
        b.y = W[(size_t)(k + kh + 1) * HC + col0 + m];
        acc = __builtin_amdgcn_wmma_f32_16x16x4_f32(false, a, false, b,
                                                    (short)0, acc, false, false);
    }
    const int rbase = (lane >> 4) * 8;
    float* __restrict__ hp = h + (size_t)(row0 + rbase) * HC + col0 + m;
    #pragma unroll
    for (int r = 0; r < 8; ++r) hp[(size_t)r * HC] = acc[r];
}

// ---------------- per-node attention halves ---------------------------------
__global__ void k_node_attn(const float* __restrict__ h,
                            const float* __restrict__ att_src,
                            const float* __restrict__ att_dst,
                            float* __restrict__ a_src, float* __restrict__ a_dst,
                            int n) {
    int i = blockIdx.x * blockDim.x + threadIdx.x;   // i = node*HEADS + head
    if (i >= n * HEADS) return;
    int node = i >> 2, head = i & 3;
    const float* __restrict__ hp = h + (size_t)node * HC + head * CPH;
    const float* __restrict__ as = att_src + head * CPH;
    const float* __restrict__ ad = att_dst + head * CPH;
    float s = 0.0f, d = 0.0f;
    #pragma unroll 8
    for (int c = 0; c < CPH; ++c) { float v = hp[c]; s += v * as[c]; d += v * ad[c]; }
    a_src[i] = s;
    a_dst[i] = d;
}

// ---------------- edge helpers ----------------------------------------------
__device__ __forceinline__ float leaky(float v) { return v > 0.0f ? v : NEG_SLOPE * v; }

// sign-split float atomic max (monotone-correct across mixed signs, -inf init)
__device__ __forceinline__ void atomicMaxF32(float* addr, float val) {
    if (val >= 0.0f) atomicMax((int*)addr, __float_as_int(val));
    else             atomicMin((unsigned int*)addr, __float_as_uint(val));
}

__device__ __forceinline__ void edge_nodes(const int* __restrict__ ei, int e, int E,
                                           int& s, int& d) {
    if (e < E) { s = ei[e]; d = ei[E + e]; }   // edge_index[0]=src, [1]=dst
    else       { s = d = e - E; }              // appended self loop
}

// ---------------- pass 1: segment max over dst ------------------------------
__global__ void k_edge_max(const int* __restrict__ ei,
                           const float* __restrict__ a_src,
                           const float* __restrict__ a_dst,
                           float* __restrict__ emax, int E, int ET) {
    int t = blockIdx.x * blockDim.x + threadIdx.x;   // t = edge*HEADS + head
    if (t >= ET * HEADS) return;
    int e = t >> 2, head = t & 3, s, d;
    edge_nodes(ei, e, E, s, d);
    float v = leaky(a_src[s * HEADS + head] + a_dst[d * HEADS + head]);
    atomicMaxF32(&emax[d * HEADS + head], v);
}

// ---------------- pass 2: segment sum of exp --------------------------------
__global__ void k_edge_denom(const int* __restrict__ ei,
                             const float* __restrict__ a_src,
                             const float* __restrict__ a_dst,
                             const float* __restrict__ emax,
                             float* __restrict__ denom, int E, int ET) {
    int t = blockIdx.x * blockDim.x + threadIdx.x;
    if (t >= ET * HEADS) return;
    int e = t >> 2, head = t & 3, s, d;
    edge_nodes(ei, e, E, s, d);
    float v = leaky(a_src[s * HEADS + head] + a_dst[d * HEADS + head]);
    atomicAdd(&denom[d * HEADS + head], __expf(v - emax[d * HEADS + head]));
}

// ---------------- pass 3: weighted scatter (one wave per edge) --------------
// lane covers channels [lane*4, lane*4+4) of 128 -> head = lane/8
__global__ __launch_bounds__(256) void k_edge_agg(const int* __restrict__ ei,
                                                  const float* __restrict__ a_src,
                                                  const float* __restrict__ a_dst,
                                                  const float* __restrict__ emax,
                                                  const float* __restrict__ denom,
                                                  const float* __restrict__ h,
                                                  float* __restrict__ agg,
                                                  int E, int ET) {
    int gwave = (blockIdx.x * blockDim.x + threadIdx.x) >> 5;
    int lane  = threadIdx.x & 31;
    if (gwave >= ET) return;
    int s, d;
    edge_nodes(ei, gwave, E, s, d);
    int head = lane >> 3;
    float e     = leaky(a_src[s * HEADS + head] + a_dst[d * HEADS + head]);
    float alpha = __expf(e - emax[d * HEADS + head]) / denom[d * HEADS + head];
    const float4 v = ((const float4*)(h + (size_t)s * HC))[lane];
    float* __restrict__ ap = agg + (size_t)d * HC + lane * 4;
    atomicAdd(ap + 0, alpha * v.x);
    atomicAdd(ap + 1, alpha * v.y);
    atomicAdd(ap + 2, alpha * v.z);
    atomicAdd(ap + 3, alpha * v.w);
}

// ---------------- GEMM2: out = elu(agg+bias) @ lin_W + lin_b  (fp32 WMMA) ---
// block = 128 threads = 4 waves; wave w -> output cols [w*16..+16), rows [b*16..+16)
__global__ __launch_bounds__(128) void k_gemm_out(const float* __restrict__ agg,
                                                  const float* __restrict__ bias,
                                                  const float* __restrict__ linW,
                                                  const float* __restrict__ linb,
                                                  float* __restrict__ out) {
    const int wave = threadIdx.x >> 5;
    const int lane = threadIdx.x & 31;
    const int row0 = blockIdx.x * 16;
    const int col0 = wave * 16;
    const int m  = lane & 15;
    const int kh = (lane >> 4) << 1;

    const float* __restrict__ arow = agg + (size_t)(row0 + m) * HC;
    v8f acc = {};
    #pragma unroll 4
    for (int k = 0; k < HC; k += 4) {
        float a0 = arow[k + kh]     + bias[k + kh];
        float a1 = arow[k + kh + 1] + bias[k + kh + 1];
        v2f a, b;
        a.x = a0 > 0.0f ? a0 : __expf(a0) - 1.0f;   // ELU (alpha=1)
        a.y = a1 > 0.0f ? a1 : __expf(a1) - 1.0f;
        b.x = linW[(k + kh)     * OUTC + col0 + m];
        b.y = linW[(k + kh + 1) * OUTC + col0 + m];
        acc = __builtin_amdgcn_wmma_f32_16x16x4_f32(false, a, false, b,
                                                    (short)0, acc, false, false);
    }
    const int rbase = (lane >> 4) * 8;
    const float lb = linb[col0 + m];
    float* __restrict__ op = out + (size_t)(row0 + rbase) * OUTC + col0 + m;
    #pragma unroll
    for (int r = 0; r < 8; ++r) op[(size_t)r * OUTC] = acc[r] + lb;
}

// ---------------- launcher ---------------------------------------------------
extern "C" void kernel_launch(void* const* d_in, const int* in_sizes, int n_in,
                              void* d_out, int out_size, void* d_ws, size_t ws_size,
                              hipStream_t stream) {
    const float* x       = (const float*)d_in[0];
    const int*   ei      = (const int*)  d_in[1];
    const float* W       = (const float*)d_in[2];
    const float* att_src = (const float*)d_in[3];
    const float* att_dst = (const float*)d_in[4];
    const float* bias    = (const float*)d_in[5];
    const float* linW    = (const float*)d_in[6];
    const float* linb    = (const float*)d_in[7];
    float* out = (float*)d_out;

    const int N  = in_sizes[0] / IN_CH;      // 50000
    const int E  = in_sizes[1] / 2;          // 1600000
    const int ET = E + N;                    // + self loops

    // workspace layout (floats)
    float* h     = (float*)d_ws;                       // N*128
    float* agg   = h     + (size_t)N * HC;             // N*128
    float* a_src = agg   + (size_t)N * HC;             // N*4
    float* a_dst = a_src + (size_t)N * HEADS;          // N*4
    float* emax  = a_dst + (size_t)N * HEADS;          // N*4
    float* denom = emax  + (size_t)N * HEADS;          // N*4

    // 0) re-init accumulators (graph replays must not see stale state)
    {
        int total = N * HC;
        k_init<<<(total + 255) / 256, 256, 0, stream>>>(agg, emax, denom, N);
    }
    // 1) h = x @ W
    k_gemm_xw<<<N / 16, 256, 0, stream>>>(x, W, h);
    // 2) per-node attention halves
    {
        int total = N * HEADS;
        k_node_attn<<<(total + 255) / 256, 256, 0, stream>>>(h, att_src, att_dst,
                                                             a_src, a_dst, N);
    }
    // 3) segment max
    {
        int total = ET * HEADS;
        k_edge_max<<<(total + 255) / 256, 256, 0, stream>>>(ei, a_src, a_dst,
                                                            emax, E, ET);
    }
    // 4) segment sum of exp
    {
        int total = ET * HEADS;
        k_edge_denom<<<(total + 255) / 256, 256, 0, stream>>>(ei, a_src, a_dst,
                                                              emax, denom, E, ET);
    }
    // 5) weighted scatter: one wave per edge
    {
        long long threads = (long long)ET * 32;
        int blocks = (int)((threads + 255) / 256);
        k_edge_agg<<<blocks, 256, 0, stream>>>(ei, a_src, a_dst, emax, denom,
                                               h, agg, E, ET);
    }
    // 6) out = elu(agg + bias) @ lin_W + lin_b
    k_gemm_out<<<N / 16, 128, 0, stream>>>(agg, bias, linW, linb, out);
}